// GCN_51058571215056
// MI455X (gfx1250) — compile-verified
//
#include <hip/hip_runtime.h>
#include <hip/hip_bf16.h>
#include <math.h>

#define N_NODES 50000
#define N_EDGES 600000
#define FEAT 128
#define HID 128
#define NGRAPH 64
#define NCLASS 2

typedef __attribute__((ext_vector_type(2))) float v2f;
typedef __attribute__((ext_vector_type(8))) float v8f;
typedef __attribute__((ext_vector_type(4))) int   v4i;

// ---- CDNA5 async global->LDS staging (guarded; falls back to sync copy) ----
#if defined(__gfx1250__) && __has_builtin(__builtin_amdgcn_global_load_async_to_lds_b128)
#define USE_ASYNC_LDS 1
#else
#define USE_ASYNC_LDS 0
#endif

typedef __attribute__((address_space(1))) v4i as1_v4i;
typedef __attribute__((address_space(3))) v4i as3_v4i;

__device__ __forceinline__ void async_copy16(const float* g, float* l) {
#if USE_ASYNC_LDS
    __builtin_amdgcn_global_load_async_to_lds_b128((as1_v4i*)g, (as3_v4i*)l, 0, 0);
#endif
}

__device__ __forceinline__ void wait_async_lds() {
#if USE_ASYNC_LDS
#if __has_builtin(__builtin_amdgcn_s_wait_asynccnt)
    __builtin_amdgcn_s_wait_asynccnt(0);
#else
    asm volatile("s_wait_asynccnt 0" ::: "memory");
#endif
#endif
}

// ---------------- degree / normalization ----------------

__global__ void k_init_deg(float* deg) {
    int i = blockIdx.x * blockDim.x + threadIdx.x;
    if (i < N_NODES) deg[i] = 1.0f;   // self-loop contributes 1 to every dst
}

__global__ void k_count_deg(const int* __restrict__ dst, float* deg) {
    int e = blockIdx.x * blockDim.x + threadIdx.x;
    if (e < N_EDGES) atomicAdd(&deg[dst[e]], 1.0f);
}

__global__ void k_dinv(float* deg) {
    int i = blockIdx.x * blockDim.x + threadIdx.x;
    if (i < N_NODES) deg[i] = rsqrtf(deg[i]);   // deg >= 1 always
}

// ---------------- fp32 WMMA GEMM: out[N,128] = A[N,128] @ W[128,128] ----------------
// One block: 16 rows. 8 waves, wave w owns output columns [16w, 16w+16).
// A tile (16x128) and full W (128x128) staged into LDS (async DMA when available).
// V_WMMA_F32_16X16X4_F32, 32 MACs of K=4 each. RELU is a compile-time flag so the
// layer-1 instance has zero overhead and layer-2 gets plain v_max (no cndmask).

template<bool RELU>
__launch_bounds__(256)
__global__ void k_gemm_wmma(const float* __restrict__ A, const float* __restrict__ Wg,
                            float* __restrict__ out) {
    __shared__ float sA[16][132];     // padded: conflict-free column reads
    __shared__ float sW[128][132];    // 128x128 weights, padded rows
    const int r0  = blockIdx.x * 16;
    const int tid = threadIdx.x;

#if USE_ASYNC_LDS
    // A tile: 16 rows x 32 float4 chunks = 512 async b128 copies
    for (int idx = tid; idx < 16 * 32; idx += 256) {
        const int row = idx >> 5;
        const int c4  = (idx & 31) << 2;
        async_copy16(A + (size_t)(r0 + row) * FEAT + c4, &sA[row][c4]);
    }
    // W: 128 rows x 32 float4 chunks = 4096 async b128 copies
    for (int idx = tid; idx < 128 * 32; idx += 256) {
        const int row = idx >> 5;
        const int c4  = (idx & 31) << 2;
        async_copy16(Wg + (size_t)row * HID + c4, &sW[row][c4]);
    }
    wait_async_lds();
#else
    for (int idx = tid; idx < 16 * 32; idx += 256) {
        const int row = idx >> 5;
        const int c4  = (idx & 31) << 2;
        float4 v = *(const float4*)(A + (size_t)(r0 + row) * FEAT + c4);
        if (RELU) {
            v.x = fmaxf(v.x, 0.f); v.y = fmaxf(v.y, 0.f);
            v.z = fmaxf(v.z, 0.f); v.w = fmaxf(v.w, 0.f);
        }
        sA[row][c4 + 0] = v.x; sA[row][c4 + 1] = v.y;
        sA[row][c4 + 2] = v.z; sA[row][c4 + 3] = v.w;
    }
    for (int idx = tid; idx < 128 * 32; idx += 256) {
        const int row = idx >> 5;
        const int c4  = (idx & 31) << 2;
        const float4 v = *(const float4*)(Wg + (size_t)row * HID + c4);
        sW[row][c4 + 0] = v.x; sW[row][c4 + 1] = v.y;
        sW[row][c4 + 2] = v.z; sW[row][c4 + 3] = v.w;
    }
#endif
    __syncthreads();

    const int wave  = tid >> 5;
    const int lane  = tid & 31;
    const int c0    = wave * 16;
    const int mn    = lane & 15;                // row (A) / col (B,C)
    const int khalf = (lane >> 4) * 2;          // K sub-offset: 0 or 2

    v8f acc = {};
#pragma unroll
    for (int k0 = 0; k0 < FEAT; k0 += 4) {
        v2f a, b;
        // A 16x4 fragment (ISA 7.12.2): lane(0-15)->K{0,1}, lane(16-31)->K{2,3}
        a.x = sA[mn][k0 + khalf];
        a.y = sA[mn][k0 + khalf + 1];
#if USE_ASYNC_LDS
        if (RELU) { a.x = fmaxf(a.x, 0.f); a.y = fmaxf(a.y, 0.f); }
#endif
        // B 4x16 fragment: same K split, N striped across lanes
        b.x = sW[k0 + khalf][c0 + mn];
        b.y = sW[k0 + khalf + 1][c0 + mn];
        acc = __builtin_amdgcn_wmma_f32_16x16x4_f32(
            false, a, false, b, (short)0, acc, false, false);
    }

    // C/D layout: VGPR r -> row (8*(lane>>4) + r), col = lane&15
    const int mb = (lane >> 4) * 8;
#pragma unroll
    for (int r = 0; r < 8; ++r)
        out[(size_t)(r0 + mb + r) * HID + c0 + mn] = acc[r];
}

// ---------------- aggregation ----------------
// out[i] = bias + t[i]*dinv[i]^2  (self-loop + bias fused), float4 vectorized
__global__ void k_agg_init(const float* __restrict__ t, const float* __restrict__ dinv,
                           const float* __restrict__ bias, float* __restrict__ out) {
    int idx4 = blockIdx.x * blockDim.x + threadIdx.x;   // over N*HID/4
    if (idx4 < N_NODES * (HID / 4)) {
        const int i  = idx4 >> 5;                       // 32 float4 per row
        const int f4 = (idx4 & 31) << 2;
        const float di = dinv[i];
        const float w  = di * di;
        const float4 v = *(const float4*)(t + (size_t)idx4 * 4);
        const float4 b = *(const float4*)(bias + f4);
        float4 o;
        o.x = b.x + v.x * w; o.y = b.y + v.y * w;
        o.z = b.z + v.z * w; o.w = b.w + v.w * w;
        *(float4*)(out + (size_t)idx4 * 4) = o;
    }
}

// One wave per edge: float4 gather of src row, 4 float atomic adds into dst row.
__launch_bounds__(256)
__global__ void k_agg_edges(const int* __restrict__ src, const int* __restrict__ dst,
                            const float* __restrict__ dinv, const float* __restrict__ t,
                            float* __restrict__ out) {
    const int lane = threadIdx.x & 31;
    const int e = blockIdx.x * 8 + (threadIdx.x >> 5);
    if (e >= N_EDGES) return;
    // prefetch a future edge's source row into cache (global_prefetch_b8)
    if (e + 64 < N_EDGES) {
        const int sp = src[e + 64];
        __builtin_prefetch(t + (size_t)sp * HID + lane * 4, 0, 1);
    }
    const int s = src[e], d = dst[e];
    const float w = dinv[s] * dinv[d];
    const float4 v = *(const float4*)(t + (size_t)s * HID + lane * 4);
    float* o = out + (size_t)d * HID + lane * 4;
    atomicAdd(o + 0, v.x * w);
    atomicAdd(o + 1, v.y * w);
    atomicAdd(o + 2, v.z * w);
    atomicAdd(o + 3, v.w * w);
}

// ---------------- global max pool ----------------

__global__ void k_gmax_init(float* g) {
    int i = blockIdx.x * blockDim.x + threadIdx.x;
    if (i < NGRAPH * HID) g[i] = -3.402823466e38f;
}

__device__ __forceinline__ void atomicMaxF(float* addr, float v) {
    // signed-max for >=0, unsigned-min for <0 (valid with -FLT_MAX init)
    if (v >= 0.f) atomicMax((int*)addr, __float_as_int(v));
    else          atomicMin((unsigned int*)addr, __float_as_uint(v));
}

__launch_bounds__(256)
__global__ void k_gmax(const int* __restrict__ batch, const float* __restrict__ h,
                       float* __restrict__ g) {
    const int lane = threadIdx.x & 31;
    const int i = blockIdx.x * 8 + (threadIdx.x >> 5);
    if (i >= N_NODES) return;
    const int b = batch[i];
    const float4 v = *(const float4*)(h + (size_t)i * HID + lane * 4);
    float* gp = g + (size_t)b * HID + lane * 4;
    atomicMaxF(gp + 0, v.x);
    atomicMaxF(gp + 1, v.y);
    atomicMaxF(gp + 2, v.z);
    atomicMaxF(gp + 3, v.w);
}

// ---------------- MLP head + softmax (single block) ----------------

__launch_bounds__(256)
__global__ void k_mlp(const float* __restrict__ g,
                      const float* __restrict__ Wl1, const float* __restrict__ bl1,
                      const float* __restrict__ Wl2, const float* __restrict__ bl2,
                      const float* __restrict__ Wo,  const float* __restrict__ bo,
                      float* __restrict__ out) {
    __shared__ float s0[NGRAPH * 128];
    __shared__ float s1[NGRAPH * 128];
    const int tid = threadIdx.x;

    for (int i = tid; i < NGRAPH * 128; i += 256) s0[i] = g[i];
    __syncthreads();

    // s1 = relu(s0 @ Wl1 + bl1)
    for (int idx = tid; idx < NGRAPH * 128; idx += 256) {
        const int row = idx >> 7, col = idx & 127;
        float acc = bl1[col];
        for (int k = 0; k < 128; ++k) acc += s0[row * 128 + k] * Wl1[k * 128 + col];
        s1[idx] = fmaxf(acc, 0.f);
    }
    __syncthreads();

    // s0 = relu(s1 @ Wl2 + bl2)
    for (int idx = tid; idx < NGRAPH * 128; idx += 256) {
        const int row = idx >> 7, col = idx & 127;
        float acc = bl2[col];
        for (int k = 0; k < 128; ++k) acc += s1[row * 128 + k] * Wl2[k * 128 + col];
        s0[idx] = fmaxf(acc, 0.f);
    }
    __syncthreads();

    // logits + row softmax
    if (tid < NGRAPH) {
        float z0 = bo[0], z1 = bo[1];
        for (int k = 0; k < 128; ++k) {
            const float v = s0[tid * 128 + k];
            z0 += v * Wo[k * NCLASS + 0];
            z1 += v * Wo[k * NCLASS + 1];
        }
        const float m = fmaxf(z0, z1);
        const float e0 = expf(z0 - m), e1 = expf(z1 - m);
        const float inv = 1.0f / (e0 + e1);
        out[tid * NCLASS + 0] = e0 * inv;
        out[tid * NCLASS + 1] = e1 * inv;
    }
}

// ---------------- launch ----------------

extern "C" void kernel_launch(void* const* d_in, const int* in_sizes, int n_in,
                              void* d_out, int out_size, void* d_ws, size_t ws_size,
                              hipStream_t stream) {
    const float* x   = (const float*)d_in[0];
    const int*   ei  = (const int*)d_in[1];
    const int*   src = ei;                 // edge_index[0]
    const int*   dst = ei + N_EDGES;       // edge_index[1]
    const int*   batch = (const int*)d_in[2];
    const float* W1  = (const float*)d_in[3];
    const float* b1  = (const float*)d_in[4];
    const float* W2  = (const float*)d_in[5];
    const float* b2  = (const float*)d_in[6];
    const float* Wl1 = (const float*)d_in[7];
    const float* bl1 = (const float*)d_in[8];
    const float* Wl2 = (const float*)d_in[9];
    const float* bl2 = (const float*)d_in[10];
    const float* Wo  = (const float*)d_in[11];
    const float* bo  = (const float*)d_in[12];
    float* out = (float*)d_out;

    float* ws   = (float*)d_ws;
    float* dinv = ws;                                       // N
    float* bufA = ws + N_NODES;                             // N*H
    float* bufB = bufA + (size_t)N_NODES * HID;             // N*H
    float* gbuf = bufB + (size_t)N_NODES * HID;             // G*H

    const int T = 256;

    // degrees -> dinv
    k_init_deg <<<(N_NODES + T - 1) / T, T, 0, stream>>>(dinv);
    k_count_deg<<<(N_EDGES + T - 1) / T, T, 0, stream>>>(dst, dinv);
    k_dinv     <<<(N_NODES + T - 1) / T, T, 0, stream>>>(dinv);

    // layer 1: t = x@W1 ; agg (+b1) ; (relu deferred into next gemm's A read)
    k_gemm_wmma<false><<<N_NODES / 16, T, 0, stream>>>(x, W1, bufA);
    k_agg_init <<<(N_NODES * (HID / 4) + T - 1) / T, T, 0, stream>>>(bufA, dinv, b1, bufB);
    k_agg_edges<<<(N_EDGES + 7) / 8, T, 0, stream>>>(src, dst, dinv, bufA, bufB);

    // layer 2: t = relu(h1)@W2 ; agg (+b2)
    k_gemm_wmma<true><<<N_NODES / 16, T, 0, stream>>>(bufB, W2, bufA);
    k_agg_init <<<(N_NODES * (HID / 4) + T - 1) / T, T, 0, stream>>>(bufA, dinv, b2, bufB);
    k_agg_edges<<<(N_EDGES + 7) / 8, T, 0, stream>>>(src, dst, dinv, bufA, bufB);

    // global max pool
    k_gmax_init<<<(NGRAPH * HID + T - 1) / T, T, 0, stream>>>(gbuf);
    k_gmax     <<<(N_NODES + 7) / 8, T, 0, stream>>>(batch, bufB, gbuf);

    // MLP head + softmax
    k_mlp<<<1, T, 0, stream>>>(gbuf, Wl1, bl1, Wl2, bl2, Wo, bo, out);
}